// MedianBlur_32049045963331
// MI455X (gfx1250) — compile-verified
//
#include <hip/hip_runtime.h>

// Problem constants from the reference: x is (32, 512, 512, 3) float32, NHWC.
#define B_      32
#define H_      512
#define W_      512
#define C_      3
#define TILE_W  128                 // output pixels per block in x
#define TILE_H  8                   // output rows per block
#define LDS_ROWS (TILE_H + 2)       // 10 (1-row halo top/bottom)
#define LDS_STRIDE 392              // floats per LDS row; row base 16B-aligned
// LDS row layout (floats): [0] pad, [1..3] left-halo pixel, [4..387] interior
// (16B-aligned, 384 floats = 128 px), [388..390] right-halo pixel, [391] pad.

#if __has_builtin(__builtin_amdgcn_global_load_async_to_lds_b32)
#define USE_ASYNC_LDS 1
#else
#define USE_ASYNC_LDS 0
#endif
#if __has_builtin(__builtin_amdgcn_global_load_async_to_lds_b128)
#define USE_ASYNC_B128 1
#else
#define USE_ASYNC_B128 0
#endif

typedef int v4i __attribute__((vector_size(16)));        // 4 x i32 = 16 B
typedef __attribute__((address_space(1))) int gas_int;   // global (device) AS
typedef __attribute__((address_space(3))) int las_int;   // LDS AS
typedef __attribute__((address_space(1))) v4i gas_v4i;   // global 16B vector
typedef __attribute__((address_space(3))) v4i las_v4i;   // LDS 16B vector

__device__ __forceinline__ float min3f(float a, float b, float c) {
  return __builtin_fminf(__builtin_fminf(a, b), c);      // -> v_min3
}
__device__ __forceinline__ float max3f(float a, float b, float c) {
  return __builtin_fmaxf(__builtin_fmaxf(a, b), c);      // -> v_max3
}
__device__ __forceinline__ float med3f(float a, float b, float c) {
  return __builtin_amdgcn_fmed3f(a, b, c);               // -> v_med3
}

__device__ __forceinline__ int clampi(int v, int lo, int hi) {
  return v < lo ? lo : (v > hi ? hi : v);
}

struct __align__(4) F3 { float x, y, z; };               // 12-byte pixel store

__global__ __launch_bounds__(256)
void median3x3_kernel(const float* __restrict__ x, float* __restrict__ out) {
  __shared__ float tile[LDS_ROWS * LDS_STRIDE];

  const int b   = blockIdx.z;
  const int h0  = blockIdx.y * TILE_H;
  const int w0  = blockIdx.x * TILE_W;
  const int tid = threadIdx.y * TILE_W + threadIdx.x;    // 0..255

  // ---- Stage interiors: 10 rows x 96 16B chunks = 960 b128 DMAs --------
  const int nchunk = LDS_ROWS * 96;
  for (int c = tid; c < nchunk; c += 256) {
    const int r = c / 96;
    const int q = c - r * 96;                            // 16B chunk in row
    const int gh = clampi(h0 + r - 1, 0, H_ - 1);
    // global base of this row's interior: 16B-aligned (w0 % 128 == 0)
    const float* src = x + (((size_t)b * H_ + gh) * W_ + w0) * C_ + 4 * q;
    const int doff = r * LDS_STRIDE + 4 + 4 * q;         // 16B-aligned in LDS
#if USE_ASYNC_LDS
#if USE_ASYNC_B128
    __builtin_amdgcn_global_load_async_to_lds_b128(
        (gas_v4i*)src, (las_v4i*)&tile[doff], 0, 0);
#else
    gas_int* g = (gas_int*)src;
    las_int* l = (las_int*)&tile[doff];
    __builtin_amdgcn_global_load_async_to_lds_b32(g, l, 0, 0);
    __builtin_amdgcn_global_load_async_to_lds_b32(g, l, 4, 0);
    __builtin_amdgcn_global_load_async_to_lds_b32(g, l, 8, 0);
    __builtin_amdgcn_global_load_async_to_lds_b32(g, l, 12, 0);
#endif
#else
    tile[doff + 0] = src[0]; tile[doff + 1] = src[1];
    tile[doff + 2] = src[2]; tile[doff + 3] = src[3];
#endif
  }

  // ---- Stage halo pixels: 10 rows x 2 sides, 3 floats each -------------
  if (tid < 2 * LDS_ROWS) {
    const int r    = tid >> 1;
    const int side = tid & 1;
    const int gh = clampi(h0 + r - 1, 0, H_ - 1);
    const int gw = side ? (w0 + TILE_W > W_ - 1 ? W_ - 1 : w0 + TILE_W)
                        : (w0 - 1 < 0 ? 0 : w0 - 1);
    const float* src = x + (((size_t)b * H_ + gh) * W_ + gw) * C_;
    const int doff = r * LDS_STRIDE + (side ? 388 : 1);
#if USE_ASYNC_LDS
    gas_int* g = (gas_int*)src;
    las_int* l = (las_int*)&tile[doff];
    __builtin_amdgcn_global_load_async_to_lds_b32(g, l, 0, 0);
    __builtin_amdgcn_global_load_async_to_lds_b32(g, l, 4, 0);
    __builtin_amdgcn_global_load_async_to_lds_b32(g, l, 8, 0);
#else
    tile[doff + 0] = src[0]; tile[doff + 1] = src[1]; tile[doff + 2] = src[2];
#endif
  }

#if USE_ASYNC_LDS
#if __has_builtin(__builtin_amdgcn_s_wait_asynccnt)
  __builtin_amdgcn_s_wait_asynccnt(0);
#else
  asm volatile("s_wait_asynccnt 0" ::: "memory");
#endif
#endif
  __syncthreads();

  // ---- Compute: each thread does a 4-row column of one pixel, 3 channels.
  // Horizontal sorted triples (lo/mi/hi) per row are reused across the 4
  // vertically-sliding 3x3 windows:  median9 = med3(max3(lo), med3(mi), min3(hi)).
  const int tx = threadIdx.x;                 // pixel within tile: 0..127
  const int ty = threadIdx.y;                 // 0..1 -> rows [ty*4, ty*4+4)

  float lo[6][3], mi[6][3], hi[6][3];
#pragma unroll
  for (int r = 0; r < 6; ++r) {
    // 9 consecutive floats: pixels w-1, w, w+1 (channels interleaved),
    // starting at float offset 1 + 3*tx (halos line up at tx==0 / tx==127).
    const float* row = &tile[(ty * 4 + r) * LDS_STRIDE + 1 + tx * 3];
    float a0 = row[0], a1 = row[1], a2 = row[2];
    float b0 = row[3], b1 = row[4], b2 = row[5];
    float c0 = row[6], c1 = row[7], c2 = row[8];
    lo[r][0] = min3f(a0, b0, c0); mi[r][0] = med3f(a0, b0, c0); hi[r][0] = max3f(a0, b0, c0);
    lo[r][1] = min3f(a1, b1, c1); mi[r][1] = med3f(a1, b1, c1); hi[r][1] = max3f(a1, b1, c1);
    lo[r][2] = min3f(a2, b2, c2); mi[r][2] = med3f(a2, b2, c2); hi[r][2] = max3f(a2, b2, c2);
  }

  const int w = w0 + tx;
#pragma unroll
  for (int i = 0; i < 4; ++i) {
    const int h = h0 + ty * 4 + i;
    F3 o;
    o.x = med3f(max3f(lo[i][0], lo[i + 1][0], lo[i + 2][0]),
                med3f(mi[i][0], mi[i + 1][0], mi[i + 2][0]),
                min3f(hi[i][0], hi[i + 1][0], hi[i + 2][0]));
    o.y = med3f(max3f(lo[i][1], lo[i + 1][1], lo[i + 2][1]),
                med3f(mi[i][1], mi[i + 1][1], mi[i + 2][1]),
                min3f(hi[i][1], hi[i + 1][1], hi[i + 2][1]));
    o.z = med3f(max3f(lo[i][2], lo[i + 1][2], lo[i + 2][2]),
                med3f(mi[i][2], mi[i + 1][2], mi[i + 2][2]),
                min3f(hi[i][2], hi[i + 1][2], hi[i + 2][2]));
    *(F3*)(out + (((size_t)b * H_ + h) * W_ + w) * C_) = o;
  }
}

extern "C" void kernel_launch(void* const* d_in, const int* in_sizes, int n_in,
                              void* d_out, int out_size, void* d_ws, size_t ws_size,
                              hipStream_t stream) {
  (void)in_sizes; (void)n_in; (void)d_ws; (void)ws_size; (void)out_size;
  const float* x = (const float*)d_in[0];
  float* out = (float*)d_out;
  dim3 block(TILE_W, 2, 1);                          // 256 threads = 8 wave32
  dim3 grid(W_ / TILE_W, H_ / TILE_H, B_);           // (4, 64, 32)
  median3x3_kernel<<<grid, block, 0, stream>>>(x, out);
}